// mHCResidual_58394375356737
// MI455X (gfx1250) — compile-verified
//
#include <hip/hip_runtime.h>

typedef __attribute__((ext_vector_type(4))) float v4f;

// ---------------------------------------------------------------------------
// Kernel 1: 50 Sinkhorn-Knopp iterations on the 2x2 W_raw. Trivial scalar
// work; one wave32, lane 0 computes, writes {a, b} to workspace.
//   a = W[0][0] + W[1][0]   (column-0 sum after final col-normalize)
//   b = W[0][1] + W[1][1]   (column-1 sum)
// ---------------------------------------------------------------------------
__global__ void sinkhorn2x2_kernel(const float* __restrict__ Wraw,
                                   float* __restrict__ ab) {
  if (threadIdx.x == 0) {
    float w00 = __builtin_fabsf(Wraw[0]) + 1e-8f;
    float w01 = __builtin_fabsf(Wraw[1]) + 1e-8f;
    float w10 = __builtin_fabsf(Wraw[2]) + 1e-8f;
    float w11 = __builtin_fabsf(Wraw[3]) + 1e-8f;
#pragma unroll 1
    for (int it = 0; it < 50; ++it) {
      // row-normalize
      float r0 = w00 + w01;
      float r1 = w10 + w11;
      w00 /= r0; w01 /= r0;
      w10 /= r1; w11 /= r1;
      // col-normalize
      float c0 = w00 + w10;
      float c1 = w01 + w11;
      w00 /= c0; w10 /= c0;
      w01 /= c1; w11 /= c1;
    }
    ab[0] = w00 + w10;
    ab[1] = w01 + w11;
  }
}

// ---------------------------------------------------------------------------
// Kernel 2: out = a*residual + b*sublayer, pure HBM-bandwidth streaming.
//  - b128 (float4) accesses, non-temporal TH (working set 384MB > 192MB L2,
//    zero reuse) on gfx1250 VMEM cache-policy bits
//  - unroll x4: 8 independent global_load_b128 in flight before first store
//  - global_prefetch of the next chunk of both read streams
//  - 256 threads/block = 8 wave32 waves; grid-stride for robustness
// ---------------------------------------------------------------------------
__launch_bounds__(256)
__global__ void axpby_stream_kernel(const float* __restrict__ res,
                                    const float* __restrict__ sub,
                                    float* __restrict__ out,
                                    const float* __restrict__ ab,
                                    long long n4, long long n) {
  const float a = ab[0];
  const float b = ab[1];

  const v4f* __restrict__ r4 = (const v4f*)res;
  const v4f* __restrict__ s4 = (const v4f*)sub;
  v4f* __restrict__ o4 = (v4f*)out;

  const long long tid    = (long long)blockIdx.x * blockDim.x + threadIdx.x;
  const long long stride = (long long)gridDim.x * blockDim.x;

  long long i = tid;
  // Main unrolled-by-4 stream: 8 loads in flight, then 4 FMAs + 4 NT stores.
  for (; i + 3 * stride < n4; i += 4 * stride) {
    // Prefetch the next 4-chunk of both read streams (gfx1250 global_prefetch;
    // speculative, never faults, no counter cost).
    __builtin_prefetch(r4 + i + 4 * stride, 0, 0);
    __builtin_prefetch(s4 + i + 4 * stride, 0, 0);

    v4f r0 = __builtin_nontemporal_load(r4 + i);
    v4f r1 = __builtin_nontemporal_load(r4 + i + stride);
    v4f r2 = __builtin_nontemporal_load(r4 + i + 2 * stride);
    v4f r3 = __builtin_nontemporal_load(r4 + i + 3 * stride);
    v4f s0 = __builtin_nontemporal_load(s4 + i);
    v4f s1 = __builtin_nontemporal_load(s4 + i + stride);
    v4f s2 = __builtin_nontemporal_load(s4 + i + 2 * stride);
    v4f s3 = __builtin_nontemporal_load(s4 + i + 3 * stride);

    v4f o0 = a * r0 + b * s0;
    v4f o1 = a * r1 + b * s1;
    v4f o2 = a * r2 + b * s2;
    v4f o3 = a * r3 + b * s3;

    __builtin_nontemporal_store(o0, o4 + i);
    __builtin_nontemporal_store(o1, o4 + i + stride);
    __builtin_nontemporal_store(o2, o4 + i + 2 * stride);
    __builtin_nontemporal_store(o3, o4 + i + 3 * stride);
  }
  // Remaining full float4s.
  for (; i < n4; i += stride) {
    v4f r = __builtin_nontemporal_load(r4 + i);
    v4f s = __builtin_nontemporal_load(s4 + i);
    __builtin_nontemporal_store(a * r + b * s, o4 + i);
  }
  // Scalar tail (n not divisible by 4): thread 0 only. (For this problem
  // n = 2*4096*4096 is divisible by 4, so this is dead but cheap insurance.)
  if (tid == 0) {
    for (long long k = n4 * 4; k < n; ++k) {
      out[k] = a * res[k] + b * sub[k];
    }
  }
}

extern "C" void kernel_launch(void* const* d_in, const int* in_sizes, int n_in,
                              void* d_out, int out_size, void* d_ws, size_t ws_size,
                              hipStream_t stream) {
  const float* residual = (const float*)d_in[0];   // (2, 4096, 4096) fp32
  const float* sublayer = (const float*)d_in[1];   // (2, 4096, 4096) fp32
  const float* Wraw     = (const float*)d_in[2];   // (2, 2) fp32
  float* out = (float*)d_out;
  float* ab  = (float*)d_ws;                       // 2 floats of scratch

  const long long n  = (long long)in_sizes[0];     // 33,554,432
  const long long n4 = n >> 2;                     // 8,388,608 float4s

  // Tiny scalar projection kernel (one wave32).
  sinkhorn2x2_kernel<<<1, 32, 0, stream>>>(Wraw, ab);

  // Streaming blend: 256 threads/block, 4 float4s per thread per input.
  const int threads = 256;
  long long want = (n4 + (long long)threads * 4 - 1) / ((long long)threads * 4);
  if (want < 1) want = 1;
  if (want > 1048576) want = 1048576;
  const int blocks = (int)want;                    // 8192 for this shape

  axpby_stream_kernel<<<blocks, threads, 0, stream>>>(
      residual, sublayer, out, ab, n4, n);
}